// InteractionArch_60971355734485
// MI455X (gfx1250) — compile-verified
//
#include <hip/hip_runtime.h>

// DLRM interaction arch: out[b] = concat(dense[b], triu(C C^T, k=1)) with
// C = [dense[b]; sparse[b,0..25]]  (27 x 128), computed via v_wmma_f32_16x16x4_f32.

typedef __attribute__((ext_vector_type(2))) float v2f;
typedef __attribute__((ext_vector_type(8))) float v8f;

#define DIMD        128
#define NFEAT       27          // 1 dense + 26 sparse
#define NPAD        32          // padded to 2 WMMA tiles
#define LDS_STRIDE  132         // floats per padded row (bank-conflict-free column reads)
#define TRI         351         // 27*26/2
#define OUT_STRIDE  (DIMD + TRI) // 479
#define WAVES_PER_BLOCK 2

__global__ __launch_bounds__(32 * WAVES_PER_BLOCK)
void dlrm_interaction_wmma(const float* __restrict__ dense,
                           const float* __restrict__ sparse,
                           float* __restrict__ out,
                           int batch)
{
    __shared__ float ldsC[WAVES_PER_BLOCK][NPAD * LDS_STRIDE];

    const int waveInBlock = threadIdx.x >> 5;
    const int lane        = threadIdx.x & 31;
    const int b           = blockIdx.x * WAVES_PER_BLOCK + waveInBlock;
    const bool active     = (b < batch);

    float* C = &ldsC[waveInBlock][0];

    // ---- Stage combined 32x128 tile into LDS (rows 27..31 zero) ----
    // Coalesced: lane l loads floats [4l, 4l+3] of each row (128-bit).
    const int col0 = lane * 4;
    if (active) {
        const float* dsrc = dense  + (size_t)b * DIMD;
        const float* ssrc = sparse + (size_t)b * 26 * DIMD;
        #pragma unroll
        for (int r = 0; r < NPAD; ++r) {
            float4 v;
            if (r == 0)          v = *(const float4*)(dsrc + col0);
            else if (r < NFEAT)  v = *(const float4*)(ssrc + (size_t)(r - 1) * DIMD + col0);
            else                 v = make_float4(0.f, 0.f, 0.f, 0.f);
            *(float4*)(C + r * LDS_STRIDE + col0) = v;
        }
    }
    __syncthreads();
    if (!active) return;

    // ---- Gram matrix G = C * C^T via WMMA f32 16x16x4, tiles (0,0),(0,1),(1,1) ----
    v8f acc00 = {};
    v8f acc01 = {};
    v8f acc11 = {};

    const int mrow  = lane & 15;    // M (for A) / N (for B) within tile
    const int khalf = lane >> 4;    // lane-group select for K

    #pragma unroll 8
    for (int k0 = 0; k0 < DIMD; k0 += 4) {
        // A 16x4 f32 layout: lanes 0-15 hold K = k0, k0+1; lanes 16-31 hold K = k0+2, k0+3.
        const int ka = k0 + 2 * khalf;
        v2f a0 = *(const v2f*)(C + mrow * LDS_STRIDE + ka);
        v2f a1 = *(const v2f*)(C + (16 + mrow) * LDS_STRIDE + ka);

        // B 4x16 f32 (B = C^T): VGPR j, lane group g holds row K = k0 + 2j + g, N = lane&15.
        const int kb = k0 + khalf;
        v2f b0, b1;
        b0.x = C[mrow * LDS_STRIDE + kb];
        b0.y = C[mrow * LDS_STRIDE + kb + 2];
        b1.x = C[(16 + mrow) * LDS_STRIDE + kb];
        b1.y = C[(16 + mrow) * LDS_STRIDE + kb + 2];

        acc00 = __builtin_amdgcn_wmma_f32_16x16x4_f32(false, a0, false, b0,
                                                      (short)0, acc00, false, false);
        acc01 = __builtin_amdgcn_wmma_f32_16x16x4_f32(false, a0, false, b1,
                                                      (short)0, acc01, false, false);
        acc11 = __builtin_amdgcn_wmma_f32_16x16x4_f32(false, a1, false, b1,
                                                      (short)0, acc11, false, false);
    }

    // ---- Epilogue ----
    float* outb = out + (size_t)b * OUT_STRIDE;

    // Copy the 128 dense features (already staged in LDS row 0).
    *(float4*)(outb + col0) = *(const float4*)(C + col0);

    // Extract strict upper triangle. D-matrix layout (16x16 f32):
    // VGPR v: lanes 0-15 -> M=v, N=lane; lanes 16-31 -> M=v+8, N=lane-16.
    float* outt = outb + DIMD;
    const int n     = (lane & 15);
    const int mbase = (lane >> 4) * 8;

    #pragma unroll
    for (int v = 0; v < 8; ++v) {
        // tile (0,0): i = mbase+v, j = n
        {
            const int i = mbase + v, j = n;
            if (i < j && j < NFEAT)
                outt[i * 26 - (i * (i - 1)) / 2 + (j - i - 1)] = acc00[v];
        }
        // tile (0,1): i = mbase+v, j = 16 + n
        {
            const int i = mbase + v, j = 16 + n;
            if (i < j && j < NFEAT)
                outt[i * 26 - (i * (i - 1)) / 2 + (j - i - 1)] = acc01[v];
        }
        // tile (1,1): i = 16 + mbase + v, j = 16 + n
        {
            const int i = 16 + mbase + v, j = 16 + n;
            if (i < j && j < NFEAT)
                outt[i * 26 - (i * (i - 1)) / 2 + (j - i - 1)] = acc11[v];
        }
    }
}

extern "C" void kernel_launch(void* const* d_in, const int* in_sizes, int n_in,
                              void* d_out, int out_size, void* d_ws, size_t ws_size,
                              hipStream_t stream)
{
    (void)n_in; (void)out_size; (void)d_ws; (void)ws_size;

    const float* dense  = (const float*)d_in[0];
    const float* sparse = (const float*)d_in[1];
    float*       out    = (float*)d_out;

    const int batch = in_sizes[0] / DIMD; // 16384

    const int blocks = (batch + WAVES_PER_BLOCK - 1) / WAVES_PER_BLOCK;
    dlrm_interaction_wmma<<<blocks, 32 * WAVES_PER_BLOCK, 0, stream>>>(dense, sparse, out, batch);
}